// LSSM_73280732005134
// MI455X (gfx1250) — compile-verified
//
#include <hip/hip_runtime.h>
#include <math.h>

// ---------------------------------------------------------------------------
// VMamba SS2D forward for MI455X (gfx1250, wave32, WMMA f32 16x16x4)
// ---------------------------------------------------------------------------
#define HW    1024
#define CIN   64
#define NST   16
#define KDIR  6
#define LTOT  65536
#define CHUNK 128
#define NCHUNK (LTOT / CHUNK)   // 512

typedef float v2f __attribute__((ext_vector_type(2)));
typedef float v8f __attribute__((ext_vector_type(8)));

__device__ __forceinline__ v8f wmma4(v2f a, v2f b, v8f c) {
  // D(16x16,f32) = A(16x4,f32) x B(4x16,f32) + C   -- chained for K=16/32/64
  return __builtin_amdgcn_wmma_f32_16x16x4_f32(false, a, false, b, (short)0, c,
                                               false, false);
}

// silu helpers with native rcp/exp/log (avoid IEEE div + libm log1p)
__device__ __forceinline__ float sigm(float x) {
  return __builtin_amdgcn_rcpf(1.f + __expf(-x));
}
__device__ __forceinline__ float softplusf(float x) {
  return x > 20.f ? x : __logf(1.f + __expf(x));
}

// map flattened index l of ordering j back to xc flat index c*1024+h*32+w
__device__ __forceinline__ int perm_index(int j, int l) {
  int c, h, w;
  switch (j) {
    case 0: c = l >> 10; h = (l >> 5) & 31; w = l & 31; break;          // (C,H,W)
    case 1: c = l >> 10; w = (l >> 5) & 31; h = l & 31; break;          // (C,W,H)
    case 2: h = l >> 11; c = (l >> 5) & 63; w = l & 31; break;          // (H,C,W)
    case 3: w = l >> 11; h = (l >> 6) & 31; c = l & 63; break;          // (W,H,C)
    case 4: h = l >> 11; w = (l >> 6) & 31; c = l & 63; break;          // (H,W,C)
    default: w = l >> 11; c = (l >> 5) & 63; h = l & 31; break;         // (W,C,H)
  }
  return (c << 10) + (h << 5) + w;
}

// ---------------- input projection: xz = x @ W_in^T, split, silu(z) --------
// grid(64 hw-tiles, 8 o-tiles), 32 threads (one wave per 16x16 tile), K=32
__global__ void k_inproj(const float* __restrict__ x, const float* __restrict__ Win,
                         float* __restrict__ xc_pre, float* __restrict__ zsilu) {
  const int hwt = blockIdx.x, ot = blockIdx.y;
  const int lane = threadIdx.x, nl = lane & 15, hf = lane >> 4;
  v8f acc = {};
#pragma unroll
  for (int kk = 0; kk < 8; ++kk) {
    const int kc0 = 4 * kk + 2 * hf;
    v2f a = {x[(hwt * 16 + nl) * 32 + kc0], x[(hwt * 16 + nl) * 32 + kc0 + 1]};
    v2f b = {Win[(ot * 16 + nl) * 32 + kc0], Win[(ot * 16 + nl) * 32 + kc0 + 1]};
    acc = wmma4(a, b, acc);
  }
#pragma unroll
  for (int v = 0; v < 8; ++v) {
    const int row = hwt * 16 + v + 8 * hf;   // hw position
    const int o = ot * 16 + nl;              // output channel 0..127
    const float val = acc[v];
    if (o < 64) xc_pre[o * HW + row] = val;                   // x1 -> (C,H,W)
    else        zsilu[row * 64 + (o - 64)] = val * sigm(val); // silu(z), (HW,C)
  }
}

// ---------------- depthwise 3x3 conv + bias + silu -------------------------
__global__ void k_conv(const float* __restrict__ xc_pre, const float* __restrict__ cw,
                       const float* __restrict__ cb, float* __restrict__ xc_act) {
  const int id = blockIdx.x * blockDim.x + threadIdx.x;  // 65536
  const int c = id >> 10, hw = id & 1023, h = hw >> 5, w = hw & 31;
  float s = cb[c];
#pragma unroll
  for (int ky = 0; ky < 3; ++ky) {
    const int hh = h + ky - 1;
    if (hh < 0 || hh > 31) continue;
#pragma unroll
    for (int kx = 0; kx < 3; ++kx) {
      const int ww = w + kx - 1;
      if (ww < 0 || ww > 31) continue;
      s += xc_pre[c * HW + hh * 32 + ww] * cw[c * 9 + ky * 3 + kx];
    }
  }
  xc_act[id] = s * sigm(s);
}

// ---------------- fused projection weight Wc[k][48][16] --------------------
// rows 0..15: dt = dtw @ xpw[:2] ; rows 16..31: B rows ; rows 32..47: C rows
__global__ void k_prepw(const float* __restrict__ xpw, const float* __restrict__ dtw,
                        float* __restrict__ Wc) {
  const int id = blockIdx.x * blockDim.x + threadIdx.x;
  if (id >= KDIR * 48 * 16) return;
  const int n = id & 15, d = (id >> 4) % 48, k = id / (48 * 16);
  float v;
  if (d < 16)
    v = dtw[(k * 16 + d) * 2 + 0] * xpw[(k * 34 + 0) * 16 + n] +
        dtw[(k * 16 + d) * 2 + 1] * xpw[(k * 34 + 1) * 16 + n];
  else
    v = xpw[(k * 34 + (d - 16 + 2)) * 16 + n];
  Wc[id] = v;
}

// ---------------- chunked selective scan (passes A and C) ------------------
// block = 256 threads = 16 dg x 16 n ; blockIdx = (chunk, k, branch)
template <bool PASSC>
__global__ void k_scan(const float* __restrict__ xc_act, const float* __restrict__ Wc,
                       const float* __restrict__ wxs, const float* __restrict__ bxs,
                       const float* __restrict__ A_logs, const float* __restrict__ dtb,
                       const float* __restrict__ Ds, const float* __restrict__ w_y,
                       const float* __restrict__ b_y,
                       float* __restrict__ localH, float* __restrict__ Ssum,
                       const float* __restrict__ Hin, float* __restrict__ outy) {
  __shared__ float sXS[16][CHUNK];  // xs (= u), row = state n / channel dg
  __shared__ float sDT[16][CHUNK];  // dt rows (pre-bias, pre-softplus)
  __shared__ float sB[16][CHUNK];   // B rows
  __shared__ float sC[16][CHUNK];   // C rows
  __shared__ float sOut[16][CHUNK]; // per-dg weighted y (pass C)
  __shared__ float sW[48][16];      // fused projection weight for this k

  const int chunk = blockIdx.x, kdir = blockIdx.y, branch = blockIdx.z;
  const int tid = threadIdx.x;
  const bool dual = (branch == 1);

  // load fused weight
  for (int e = tid; e < 48 * 16; e += 256)
    sW[e >> 4][e & 15] = Wc[(kdir * 48 + (e >> 4)) * 16 + (e & 15)];

  // gather xs slab (16 states x CHUNK columns) from xc_act on the fly
  for (int e = tid; e < 16 * CHUNK; e += 256) {
    const int n = e >> 7, col = e & (CHUNK - 1);
    const int l = chunk * CHUNK + col;
    int j, m, lpos;
    if (dual) { j = kdir; m = n; lpos = (LTOT - 1) - l; }
    else { const int idx = kdir * 16 + n; j = idx % 6; m = idx / 6; lpos = l; }
    sXS[n][col] = xc_act[perm_index(j, lpos)] * wxs[m] + bxs[m];
  }
  __syncthreads();

  // WMMA projection: [dt;B;C](48x16) @ xs(16x16) per l-tile, one wave each
  {
    const int wave = tid >> 5;                 // 8 waves = 8 l-tiles (CHUNK/16)
    const int lane = tid & 31, nl = lane & 15, hf = lane >> 4;
#pragma unroll
    for (int dt3 = 0; dt3 < 3; ++dt3) {
      v8f acc = {};
#pragma unroll
      for (int kk = 0; kk < 4; ++kk) {
        const int kc0 = 4 * kk + 2 * hf;
        v2f a = {sW[dt3 * 16 + nl][kc0], sW[dt3 * 16 + nl][kc0 + 1]};
        v2f b = {sXS[kc0][wave * 16 + nl], sXS[kc0 + 1][wave * 16 + nl]};
        acc = wmma4(a, b, acc);
      }
#pragma unroll
      for (int v = 0; v < 8; ++v) {
        const int row = v + 8 * hf, col = wave * 16 + nl;
        if (dt3 == 0)      sDT[row][col] = acc[v];
        else if (dt3 == 1) sB[row][col] = acc[v];
        else               sC[row][col] = acc[v];
      }
    }
  }
  __syncthreads();

  // sequential recurrence within chunk; thread (dg,n) owns one state scalar
  const int dg = tid >> 4, n = tid & 15;
  const int ch = kdir * 16 + dg;
  const float Aval = -__expf(A_logs[ch * 16 + n]);
  const float bias = dtb[ch];
  const float Dv = Ds[ch];
  const float wy = w_y[dg];
  const size_t blk = (size_t)(branch * 6 + kdir);

  float h = PASSC ? Hin[(blk * NCHUNK + chunk) * 256 + tid] : 0.f;
  float sdel = 0.f;
  for (int l = 0; l < CHUNK; ++l) {
    const float dl = softplusf(sDT[dg][l] + bias);
    const float u = sXS[dg][l];
    h = h * __expf(dl * Aval) + dl * u * sB[n][l];
    if (PASSC) {
      float yp = h * sC[n][l];
      yp += __shfl_xor(yp, 8, 16);
      yp += __shfl_xor(yp, 4, 16);
      yp += __shfl_xor(yp, 2, 16);
      yp += __shfl_xor(yp, 1, 16);
      if (n == 0) sOut[dg][l] = wy * (yp + u * Dv);
    } else {
      sdel += dl;
    }
  }

  if (PASSC) {
    __syncthreads();
    if (tid < CHUNK) {
      float s = 0.f;
#pragma unroll
      for (int d2 = 0; d2 < 16; ++d2) s += sOut[d2][tid];
      outy[blk * LTOT + chunk * CHUNK + tid] = s + b_y[0];
    }
  } else {
    localH[(blk * NCHUNK + chunk) * 256 + tid] = h;
    if (n == 0) Ssum[(blk * NCHUNK + chunk) * 16 + dg] = sdel;
  }
}

// ---------------- pass B: cross-chunk state combine ------------------------
__global__ void k_combine(const float* __restrict__ localH, const float* __restrict__ Ssum,
                          const float* __restrict__ A_logs, float* __restrict__ Hin) {
  const int blk = blockIdx.x;  // branch*6 + k
  const int k = blk % 6;
  const int t = threadIdx.x, dg = t >> 4, n = t & 15;
  const float Aval = -__expf(A_logs[(k * 16 + dg) * 16 + n]);
  float Hp = 0.f;
  for (int c = 0; c < NCHUNK; ++c) {
    const size_t idx = ((size_t)blk * NCHUNK + c) * 256 + t;
    Hin[idx] = Hp;  // state entering chunk c
    const float S = Ssum[((size_t)blk * NCHUNK + c) * 16 + dg];
    Hp = __expf(S * Aval) * Hp + localH[idx];
  }
}

// ---------------- gather: inverse-permute 6 directions, sum ----------------
__global__ void k_gather(const float* __restrict__ outy, float* __restrict__ ytot) {
  const int id = blockIdx.x * blockDim.x + threadIdx.x;  // 131072
  const int b = id >> 16, rem = id & 65535;
  const int c = rem >> 10, hw = rem & 1023, h = hw >> 5, w = hw & 31;
  const int l[6] = {(c << 10) + (h << 5) + w,  (c << 10) + (w << 5) + h,
                    (h << 11) + (c << 5) + w,  (w << 11) + (h << 6) + c,
                    (h << 11) + (w << 6) + c,  (w << 11) + (c << 5) + h};
  float s = 0.f;
#pragma unroll
  for (int j = 0; j < 6; ++j) {
    const int lj = (b == 1) ? (LTOT - 1 - l[j]) : l[j];
    s += outy[((size_t)b * 6 + j) * LTOT + lj];
  }
  ytot[(size_t)b * LTOT + hw * 64 + c] = s;
}

// ---------------- layernorm(both branches) * silu(z) -----------------------
__global__ void k_norm(const float* __restrict__ ytot, const float* __restrict__ zsilu,
                       const float* __restrict__ lnw, const float* __restrict__ lnb,
                       float* __restrict__ g) {
  const int hw = blockIdx.x, lane = threadIdx.x;  // 32 threads, 2 channels each
  float res0 = 0.f, res1 = 0.f;
#pragma unroll
  for (int br = 0; br < 2; ++br) {
    const float* row = ytot + (size_t)br * LTOT + hw * 64;
    const float v0 = row[lane], v1 = row[lane + 32];
    float s = v0 + v1, q = v0 * v0 + v1 * v1;
#pragma unroll
    for (int off = 16; off; off >>= 1) {
      s += __shfl_xor(s, off, 32);
      q += __shfl_xor(q, off, 32);
    }
    const float mean = s * (1.f / 64.f);
    const float var = q * (1.f / 64.f) - mean * mean;
    const float inv = rsqrtf(var + 1e-5f);
    res0 += (v0 - mean) * inv * lnw[lane] + lnb[lane];
    res1 += (v1 - mean) * inv * lnw[lane + 32] + lnb[lane + 32];
  }
  g[hw * 64 + lane]      = res0 * zsilu[hw * 64 + lane];
  g[hw * 64 + lane + 32] = res1 * zsilu[hw * 64 + lane + 32];
}

// ---------------- output projection: g(1024x64) @ W_out^T(64x32) -----------
__global__ void k_outproj(const float* __restrict__ g, const float* __restrict__ Wout,
                          float* __restrict__ out) {
  const int hwt = blockIdx.x, ot = blockIdx.y;
  const int lane = threadIdx.x, nl = lane & 15, hf = lane >> 4;
  v8f acc = {};
#pragma unroll
  for (int kk = 0; kk < 16; ++kk) {
    const int kc0 = 4 * kk + 2 * hf;
    v2f a = {g[(hwt * 16 + nl) * 64 + kc0], g[(hwt * 16 + nl) * 64 + kc0 + 1]};
    v2f b = {Wout[(ot * 16 + nl) * 64 + kc0], Wout[(ot * 16 + nl) * 64 + kc0 + 1]};
    acc = wmma4(a, b, acc);
  }
#pragma unroll
  for (int v = 0; v < 8; ++v)
    out[(hwt * 16 + v + 8 * hf) * 32 + ot * 16 + nl] = acc[v];
}

// ---------------------------------------------------------------------------
extern "C" void kernel_launch(void* const* d_in, const int* in_sizes, int n_in,
                              void* d_out, int out_size, void* d_ws, size_t ws_size,
                              hipStream_t stream) {
  const float* x    = (const float*)d_in[0];
  const float* Win  = (const float*)d_in[1];
  const float* cw   = (const float*)d_in[2];
  const float* cb   = (const float*)d_in[3];
  const float* wxs  = (const float*)d_in[4];
  const float* bxs  = (const float*)d_in[5];
  const float* xpw  = (const float*)d_in[6];
  const float* dtw  = (const float*)d_in[7];
  const float* dtb  = (const float*)d_in[8];
  const float* Alog = (const float*)d_in[9];
  const float* Dsv  = (const float*)d_in[10];
  const float* wy   = (const float*)d_in[11];
  const float* by   = (const float*)d_in[12];
  const float* lnw  = (const float*)d_in[13];
  const float* lnb  = (const float*)d_in[14];
  const float* Wout = (const float*)d_in[15];
  float* out = (float*)d_out;
  float* ws = (float*)d_ws;

  size_t o = 0;
  float* xc_pre = ws + o; o += LTOT;                      // (C,H,W) pre-conv
  float* xc_act = ws + o; o += LTOT;                      // (C,H,W) post conv+silu
  float* zsilu  = ws + o; o += LTOT;                      // (HW,C)
  float* Wc     = ws + o; o += KDIR * 48 * 16;            // fused proj weights
  float* localH = ws + o; o += (size_t)12 * NCHUNK * 256; // per-chunk final states
  float* Ssum   = ws + o; o += (size_t)12 * NCHUNK * 16;  // per-chunk delta sums
  float* Hin    = ws + o; o += (size_t)12 * NCHUNK * 256; // per-chunk incoming states
  float* outy   = ws + o; o += (size_t)12 * LTOT;         // out_y per (branch,k)
  float* ytot   = ws + o; o += (size_t)2 * LTOT;          // y per branch, (HW,C)
  float* gbuf   = ws + o; o += LTOT;                      // normed*silu(z)

  k_inproj<<<dim3(64, 8), 32, 0, stream>>>(x, Win, xc_pre, zsilu);
  k_conv<<<256, 256, 0, stream>>>(xc_pre, cw, cb, xc_act);
  k_prepw<<<18, 256, 0, stream>>>(xpw, dtw, Wc);
  k_scan<false><<<dim3(NCHUNK, 6, 2), 256, 0, stream>>>(
      xc_act, Wc, wxs, bxs, Alog, dtb, Dsv, wy, by, localH, Ssum, nullptr, nullptr);
  k_combine<<<12, 256, 0, stream>>>(localH, Ssum, Alog, Hin);
  k_scan<true><<<dim3(NCHUNK, 6, 2), 256, 0, stream>>>(
      xc_act, Wc, wxs, bxs, Alog, dtb, Dsv, wy, by, localH, Ssum, Hin, outy);
  k_gather<<<512, 256, 0, stream>>>(outy, ytot);
  k_norm<<<1024, 32, 0, stream>>>(ytot, zsilu, lnw, lnb, gbuf);
  k_outproj<<<dim3(64, 2), 32, 0, stream>>>(gbuf, Wout, out);
}